// MHDA_74929999446389
// MI455X (gfx1250) — compile-verified
//
#include <hip/hip_runtime.h>

typedef __attribute__((ext_vector_type(2))) float v2f;
typedef __attribute__((ext_vector_type(8))) float v8f;

#define NTOK   2304          // 48*48 tokens
#define DMODEL 256
#define DM2    512           // 2*d_model
#define NHEAD  8
#define QTILES (NTOK / 16)   // 144
#define SCALING 0.17677669529663687f   // 1/sqrt(32)

// D = A(16x4 f32) * B(4x16 f32) + C(16x16 f32), full fp32 WMMA
#define WMMA4(A, B, C) \
  __builtin_amdgcn_wmma_f32_16x16x4_f32(false, (A), false, (B), (short)0, (C), false, false)

__device__ __forceinline__ float rmax16(float v) {
#pragma unroll
  for (int m = 1; m < 16; m <<= 1) v = fmaxf(v, __shfl_xor(v, m, 32));
  return v;
}
__device__ __forceinline__ float rsum16(float v) {
#pragma unroll
  for (int m = 1; m < 16; m <<= 1) v += __shfl_xor(v, m, 32);
  return v;
}

// ---------------------------------------------------------------------------
// Kernel 1: QKV projection.  Y[n][e] = sum_c X[c][n] * W[e][c]
// X is (C=256, N=2304) row-major;  W is (512, 256) row-major;  Y is (N, 512).
// One wave computes one 16x16 output tile; K-loop = 256/4 = 64 WMMAs.
// gridDim.z selects Wq / Wk / Wv.
// ---------------------------------------------------------------------------
__global__ void __launch_bounds__(256) proj_kernel(
    const float* __restrict__ X,
    const float* __restrict__ Wq, const float* __restrict__ Wk, const float* __restrict__ Wv,
    float* __restrict__ Qb, float* __restrict__ Kb, float* __restrict__ Vb) {
  const float* W = (blockIdx.z == 0) ? Wq : ((blockIdx.z == 1) ? Wk : Wv);
  float* Y       = (blockIdx.z == 0) ? Qb : ((blockIdx.z == 1) ? Kb : Vb);
  int wave = threadIdx.x >> 5, lane = threadIdx.x & 31;
  int g = lane >> 4, r = lane & 15;
  int wid = blockIdx.x * 8 + wave;   // 0..4607
  int mt = wid >> 5;                 // token tile 0..143
  int et = wid & 31;                 // feature tile 0..31
  int m0 = mt * 16, e0 = et * 16;

  const float* wrow = W + (size_t)(e0 + r) * DMODEL;
  v8f acc = {};
#pragma unroll 8
  for (int ks = 0; ks < 64; ++ks) {
    int k = ks * 4 + 2 * g;
    v2f a = { X[(size_t)k * NTOK + m0 + r], X[(size_t)(k + 1) * NTOK + m0 + r] };
    v2f b = { wrow[k], wrow[k + 1] };
    acc = WMMA4(a, b, acc);
  }
#pragma unroll
  for (int i = 0; i < 8; ++i)
    Y[(size_t)(m0 + i + 8 * g) * DM2 + e0 + r] = acc[i];
}

// ---------------------------------------------------------------------------
// Kernel 2: fused differential flash attention + RMS-norm epilogue.
// One wave owns one (head, 16-row query tile); causal loop over kv tiles.
// Two online softmaxes (S1, S2) accumulate O1, O2 against the shared V;
// final O = O1/L1 - lam * O2/L2, then per-(token,head) RMS norm * 0.2 * scale.
// P fragments are transposed C-layout -> A-layout through per-wave LDS.
// ---------------------------------------------------------------------------
__global__ void __launch_bounds__(128) attn_kernel(
    const float* __restrict__ Q, const float* __restrict__ K, const float* __restrict__ V,
    float* __restrict__ On,
    const float* __restrict__ lq1, const float* __restrict__ lk1,
    const float* __restrict__ lq2, const float* __restrict__ lk2,
    const float* __restrict__ rms_scale) {
  __shared__ float ldsP[4][2][256];   // [wave][P1/P2][16x16]
  int wave = threadIdx.x >> 5, lane = threadIdx.x & 31;
  int g = lane >> 4, r = lane & 15;
  int wid = blockIdx.x * 4 + wave;    // 0..1151
  int h = wid / QTILES;
  int qt = wid % QTILES;
  int m0 = qt * 16;
  int qoff = h * 64;
  float* P1 = ldsP[wave][0];
  float* P2 = ldsP[wave][1];

  // lambda = exp(lq1.lk1) - exp(lq2.lk2) + 0.8   (per head, 32-dim dots)
  float lp1 = lq1[h * 32 + lane] * lk1[h * 32 + lane];
  float lp2 = lq2[h * 32 + lane] * lk2[h * 32 + lane];
#pragma unroll
  for (int m = 1; m < 32; m <<= 1) {
    lp1 += __shfl_xor(lp1, m, 32);
    lp2 += __shfl_xor(lp2, m, 32);
  }
  float lam = __expf(lp1) - __expf(lp2) + 0.8f;

  // preload Q1/Q2 A-fragments (reused every kv tile)
  const float* qrow = Q + (size_t)(m0 + r) * DM2 + qoff;
  v2f qa1[8], qa2[8];
#pragma unroll
  for (int ks = 0; ks < 8; ++ks) {
    int k = ks * 4 + 2 * g;
    qa1[ks] = v2f{ qrow[k], qrow[k + 1] };
    qa2[ks] = v2f{ qrow[32 + k], qrow[32 + k + 1] };
  }

  v8f o1[4] = {}, o2[4] = {};
  float M1[8], L1[8], M2[8], L2[8];
#pragma unroll
  for (int i = 0; i < 8; ++i) { M1[i] = M2[i] = -1e30f; L1[i] = L2[i] = 0.f; }

  for (int kv = 0; kv <= qt; ++kv) {
    int n0 = kv * 16;
    const float* krow = K + (size_t)(n0 + r) * DM2 + qoff;
    if (kv < qt)  // warm next K tile through the cache hierarchy
      __builtin_prefetch((const void*)(K + (size_t)(n0 + 16 + r) * DM2 + qoff), 0, 1);

    // S1 = Q1*K1^T, S2 = Q2*K2^T  (16 WMMAs)
    v8f s1 = {}, s2 = {};
#pragma unroll
    for (int ks = 0; ks < 8; ++ks) {
      int k = ks * 4 + 2 * g;
      v2f b1 = { krow[k], krow[k + 1] };
      s1 = WMMA4(qa1[ks], b1, s1);
      v2f b2 = { krow[32 + k], krow[32 + k + 1] };
      s2 = WMMA4(qa2[ks], b2, s2);
    }

    bool diag = (kv == qt);
    float f1[8], f2[8];
#pragma unroll
    for (int i = 0; i < 8; ++i) {
      float v1 = s1[i] * SCALING, v2 = s2[i] * SCALING;
      if (diag && (r > i + 8 * g)) { v1 = -1e30f; v2 = -1e30f; }   // causal mask
      float nM1 = fmaxf(M1[i], rmax16(v1));
      float nM2 = fmaxf(M2[i], rmax16(v2));
      float p1 = __expf(v1 - nM1);
      float p2 = __expf(v2 - nM2);
      f1[i] = __expf(M1[i] - nM1);
      f2[i] = __expf(M2[i] - nM2);
      M1[i] = nM1; M2[i] = nM2;
      L1[i] = L1[i] * f1[i] + rsum16(p1);
      L2[i] = L2[i] * f2[i] + rsum16(p2);
      P1[(i + 8 * g) * 16 + r] = p1;    // C-layout -> LDS (per-wave region)
      P2[(i + 8 * g) * 16 + r] = p2;
    }
#pragma unroll
    for (int j = 0; j < 4; ++j)
#pragma unroll
      for (int i = 0; i < 8; ++i) { o1[j][i] *= f1[i]; o2[j][i] *= f2[i]; }

    // reload P as A-fragments (in-wave LDS ops are ordered; no barrier needed)
    v2f pa1[4], pa2[4];
#pragma unroll
    for (int ks = 0; ks < 4; ++ks) {
      int k = ks * 4 + 2 * g;
      pa1[ks] = v2f{ P1[r * 16 + k], P1[r * 16 + k + 1] };
      pa2[ks] = v2f{ P2[r * 16 + k], P2[r * 16 + k + 1] };
    }

    // O1 += P1*V, O2 += P2*V  (32 WMMAs; V fragments shared by both)
#pragma unroll
    for (int j = 0; j < 4; ++j) {
#pragma unroll
      for (int ks = 0; ks < 4; ++ks) {
        int kk = ks * 4 + 2 * g;
        const float* vp = V + (size_t)(n0 + kk) * DM2 + qoff + j * 16 + r;
        v2f vb = { vp[0], vp[DM2] };
        o1[j] = WMMA4(pa1[ks], vb, o1[j]);
        o2[j] = WMMA4(pa2[ks], vb, o2[j]);
      }
    }
  }

  // epilogue: combine branches, RMS-norm over the 64 head dims, scale, store
  float inv1[8], inv2[8];
#pragma unroll
  for (int i = 0; i < 8; ++i) { inv1[i] = 1.f / L1[i]; inv2[i] = lam / L2[i]; }
  float rs[4];
#pragma unroll
  for (int j = 0; j < 4; ++j) rs[j] = rms_scale[j * 16 + r];

  float ov[4][8];
#pragma unroll
  for (int j = 0; j < 4; ++j)
#pragma unroll
    for (int i = 0; i < 8; ++i) ov[j][i] = o1[j][i] * inv1[i] - o2[j][i] * inv2[i];

#pragma unroll
  for (int i = 0; i < 8; ++i) {
    float ss = 0.f;
#pragma unroll
    for (int j = 0; j < 4; ++j) ss += ov[j][i] * ov[j][i];
    ss = rsum16(ss);
    float rinv = 0.2f * __frsqrt_rn(ss * (1.0f / 64.0f) + 1e-5f);  // (1-0.8)/rms
#pragma unroll
    for (int j = 0; j < 4; ++j)
      On[(size_t)(m0 + i + 8 * g) * DM2 + qoff + j * 16 + r] = ov[j][i] * rinv * rs[j];
  }
}

// ---------------------------------------------------------------------------
// Kernel 3: output projection, transposed store.
// out[d][n] = sum_e Wo[d][e] * On[n][e];  out is (C=256, N=2304) row-major.
// One wave per 16x16 tile; K-loop = 512/4 = 128 WMMAs.
// ---------------------------------------------------------------------------
__global__ void __launch_bounds__(256) oproj_kernel(
    const float* __restrict__ On, const float* __restrict__ Wo,
    float* __restrict__ out) {
  int wave = threadIdx.x >> 5, lane = threadIdx.x & 31;
  int g = lane >> 4, r = lane & 15;
  int wid = blockIdx.x * 8 + wave;   // 0..2303
  int dt = wid / QTILES;             // 0..15
  int nt = wid % QTILES;             // 0..143
  int d0 = dt * 16, n0 = nt * 16;

  const float* arow = Wo + (size_t)(d0 + r) * DM2;
  const float* brow = On + (size_t)(n0 + r) * DM2;
  v8f acc = {};
#pragma unroll 8
  for (int ks = 0; ks < 128; ++ks) {
    int k = ks * 4 + 2 * g;
    v2f a = { arow[k], arow[k + 1] };
    v2f b = { brow[k], brow[k + 1] };
    acc = WMMA4(a, b, acc);
  }
#pragma unroll
  for (int i = 0; i < 8; ++i)
    out[(size_t)(d0 + i + 8 * g) * NTOK + n0 + r] = acc[i];
}

// ---------------------------------------------------------------------------
extern "C" void kernel_launch(void* const* d_in, const int* in_sizes, int n_in,
                              void* d_out, int out_size, void* d_ws, size_t ws_size,
                              hipStream_t stream) {
  const float* X   = (const float*)d_in[0];
  const float* Wq  = (const float*)d_in[1];
  const float* Wk  = (const float*)d_in[2];
  const float* Wv  = (const float*)d_in[3];
  const float* Wo  = (const float*)d_in[4];
  const float* lq1 = (const float*)d_in[5];
  const float* lk1 = (const float*)d_in[6];
  const float* lq2 = (const float*)d_in[7];
  const float* lk2 = (const float*)d_in[8];
  const float* rms = (const float*)d_in[9];
  float* out = (float*)d_out;

  float* Qb = (float*)d_ws;                      // N x 512
  float* Kb = Qb + (size_t)NTOK * DM2;           // N x 512
  float* Vb = Kb + (size_t)NTOK * DM2;           // N x 512
  float* Ob = Vb + (size_t)NTOK * DM2;           // N x 512 (normed attention out)

  // 1) QKV projections: 144*32 tiles * 3 matrices, 8 waves/block
  dim3 pgrid(576, 1, 3);
  proj_kernel<<<pgrid, 256, 0, stream>>>(X, Wq, Wk, Wv, Qb, Kb, Vb);

  // 2) fused differential flash attention: 8 heads * 144 q-tiles, 4 waves/block
  attn_kernel<<<288, 128, 0, stream>>>(Qb, Kb, Vb, Ob, lq1, lk1, lq2, lk2, rms);

  // 3) output projection (transposed store): 16*144 tiles, 8 waves/block
  oproj_kernel<<<288, 256, 0, stream>>>(Ob, Wo, out);
}